// SincConv1d_90211493085796
// MI455X (gfx1250) — compile-verified
//
#include <hip/hip_runtime.h>
#include <math.h>

// ---------------------------------------------------------------------------
// SincConv1d for MI455X (gfx1250, wave32, WMMA)
//   out[n, oc, t] = sum_k F[oc, k] * wave[n, t + k]
//   N = 64, OC = 64, KD = 129, T_in = 16384, T_out = 16256
// Per-batch GEMM  Out[64 x 16256] = F[64 x 132] * Toeplitz(W)[132 x 16256]
// with exact-f32 V_WMMA_F32_16X16X4_F32; B built on the fly from a sliding
// LDS window. 2 batches per block to halve filter L2->LDS re-staging and
// double matrix work per barrier. Filter staging uses the CDNA5 async
// global->LDS DMA path (GLOBAL_LOAD_ASYNC_TO_LDS_B128, ASYNCcnt).
// ---------------------------------------------------------------------------

#define OC      64
#define KD      129
#define KP      132              // KD padded to multiple of 4 (zeros in 129..131)
#define TIN     16384
#define TOUT    (TIN - KD + 1)   // 16256 = 127 * 128
#define TILE_T  128
#define BLOCK   256              // 8 wave32 per block
#define NB      2                // batches per block
#define SW_LEN  264              // >= TILE_T + KP (260), padded stride

#define TWO_PI  6.28318530717958647692f
#define MIN_LOW_NORM  0.03125f   // MIN_LOW_HZ  / FS = 4/128
#define MIN_BAND_NORM 0.015625f  // MIN_BAND_HZ / FS = 2/128

typedef __attribute__((ext_vector_type(2))) float v2f;
typedef __attribute__((ext_vector_type(8))) float v8f;

// async-DMA pointer types: builtin wants vector_size(16) int pointers
typedef int v4i __attribute__((vector_size(16)));
typedef __attribute__((address_space(1))) v4i* gv4i_p;   // global
typedef __attribute__((address_space(3))) v4i* lv4i_p;   // LDS

#define HAS_ASYNC_LDS __has_builtin(__builtin_amdgcn_global_load_async_to_lds_b128)

// ---------------------------------------------------------------------------
// Phase 1: build the 64 x KP padded filter bank in workspace (f32).
// ---------------------------------------------------------------------------
__global__ __launch_bounds__(64) void build_filters_kernel(
    const float* __restrict__ low_hz,
    const float* __restrict__ band_hz,
    float* __restrict__ filt)     // [OC * KP]
{
    const int ch = threadIdx.x;
    if (ch >= OC) return;

    const float fb = fabsf(low_hz[ch]) + MIN_LOW_NORM;
    const float fe = fb + fabsf(band_hz[ch]) + MIN_BAND_NORM;

    float mx = -3.4e38f;
    for (int j = 0; j < KD; ++j) {
        float bp;
        if (j == 64) {
            bp = 2.0f * (fe - fb);
        } else {
            const float d  = fabsf((float)(j - 64));
            const float a1 = TWO_PI * fb * d;
            const float a2 = TWO_PI * fe * d;
            bp = 2.0f * fe * (sinf(a2) / a2) - 2.0f * fb * (sinf(a1) / a1);
        }
        mx = fmaxf(mx, bp);
    }
    const float inv_mx = 1.0f / mx;

    for (int j = 0; j < KD; ++j) {
        float bp;
        if (j == 64) {
            bp = 2.0f * (fe - fb);
        } else {
            const float d  = fabsf((float)(j - 64));
            const float a1 = TWO_PI * fb * d;
            const float a2 = TWO_PI * fe * d;
            bp = 2.0f * fe * (sinf(a2) / a2) - 2.0f * fb * (sinf(a1) / a1);
        }
        const float win = 0.54f - 0.46f * cosf(TWO_PI * (float)j / (float)KD);
        filt[ch * KP + j] = bp * inv_mx * win;
    }
    filt[ch * KP + 129] = 0.0f;
    filt[ch * KP + 130] = 0.0f;
    filt[ch * KP + 131] = 0.0f;
}

// ---------------------------------------------------------------------------
// Phase 2: WMMA conv. Block = (64 oc) x (128 t) x (2 batches).
// Wave wv: batch b = wv>>2, t-subtiles (wv&3) and (wv&3)+4; 4 oc-subtiles.
// K-loop: 33 steps, 8 V_WMMA_F32_16X16X4_F32 per step (A reused for 2 B's).
//
// f32 WMMA fragment layouts (ISA 7.12.2):
//   A 16x4 : lanes 0-15 row M=l16 hold K={0,1}; lanes 16-31 hold K={2,3}
//   B 4x16 : lanes 0-15 col N=l16 hold K={0,1}; lanes 16-31 hold K={2,3}
//   C/D    : lane l16 col N=l16; VGPR r = row M = r + 8*half
// ---------------------------------------------------------------------------
__global__ __launch_bounds__(BLOCK) void sinc_conv_wmma_kernel(
    const float* __restrict__ wave,   // [64 * TIN]
    const float* __restrict__ filt,   // [OC * KP]
    float* __restrict__ out)          // [64 * OC * TOUT]
{
    __shared__ __align__(16) float sf[OC * KP];      // 33792 B
    __shared__ __align__(16) float sw[NB][SW_LEN];   // 2112 B

    const int tid    = threadIdx.x;
    const int batch0 = blockIdx.y * NB;
    const int t0     = blockIdx.x * TILE_T;

    // --- stage filter bank (L2-resident) into LDS ---
#if HAS_ASYNC_LDS
    for (int i = tid * 4; i < OC * KP; i += BLOCK * 4) {
        __builtin_amdgcn_global_load_async_to_lds_b128(
            (gv4i_p)(filt + i), (lv4i_p)(&sf[i]), 0, 0);
    }
#else
    for (int i = tid * 4; i < OC * KP; i += BLOCK * 4)
        *reinterpret_cast<float4*>(&sf[i]) =
            *reinterpret_cast<const float4*>(&filt[i]);
#endif

    // --- stage 2 waveform windows (guarded: tail reads past T_in are zero) ---
    for (int i = tid; i < NB * SW_LEN; i += BLOCK) {
        const int b = i / SW_LEN;
        const int j = i - b * SW_LEN;
        const int g = t0 + j;
        sw[b][j] = (g < TIN) ? wave[(size_t)(batch0 + b) * TIN + g] : 0.0f;
    }

#if HAS_ASYNC_LDS
#if __has_builtin(__builtin_amdgcn_s_wait_asynccnt)
    __builtin_amdgcn_s_wait_asynccnt(0);
#else
    asm volatile("s_wait_asynccnt 0" ::: "memory");
#endif
#endif
    __syncthreads();

    const int wv    = tid >> 5;        // wave id 0..7
    const int b     = wv >> 2;         // local batch 0/1
    const int ts    = wv & 3;          // first t-subtile
    const int lane  = tid & 31;
    const int half  = lane >> 4;
    const int l16   = lane & 15;
    const int nloc0 = ts * 16 + l16;   // columns of the two t-subtiles
    const int nloc1 = nloc0 + 64;
    const float* __restrict__ swb = sw[b];

    v8f acc[2][4] = {};                // [t-subtile][oc-subtile]

    for (int kk = 0; kk < KP / 4; ++kk) {
        const int kb = kk * 4 + half * 2;   // this half-wave's K pair

        v2f b0, b1;                         // implicit-im2col B fragments
        b0.x = swb[nloc0 + kb];
        b0.y = swb[nloc0 + kb + 1];
        b1.x = swb[nloc1 + kb];
        b1.y = swb[nloc1 + kb + 1];

        #pragma unroll
        for (int m = 0; m < 4; ++m) {
            const v2f a = *reinterpret_cast<const v2f*>(&sf[(m * 16 + l16) * KP + kb]);
            acc[0][m] = __builtin_amdgcn_wmma_f32_16x16x4_f32(
                false, a, false, b0, (short)0, acc[0][m], false, false);
            acc[1][m] = __builtin_amdgcn_wmma_f32_16x16x4_f32(
                false, a, false, b1, (short)0, acc[1][m], false, false);
        }
    }

    // --- store D fragments: lane l16 -> column, VGPR r -> row M = r + 8*half ---
    const size_t obase = (size_t)(batch0 + b) * OC;
    #pragma unroll
    for (int s = 0; s < 2; ++s) {
        const int tcol = t0 + (s == 0 ? nloc0 : nloc1);
        #pragma unroll
        for (int m = 0; m < 4; ++m) {
            const int oc0 = m * 16 + half * 8;
            float* op = out + (obase + oc0) * TOUT + tcol;
            #pragma unroll
            for (int r = 0; r < 8; ++r)
                op[(size_t)r * TOUT] = acc[s][m][r];
        }
    }
}

// ---------------------------------------------------------------------------
// Host entry
//   d_in[0] waveforms f32 [64*1*16384]
//   d_in[1] low_hz_   f32 [64]
//   d_in[2] band_hz_  f32 [64]
//   d_out   f32 [64*64*16256]
//   d_ws    >= OC*KP*4 = 33792 bytes (padded filter bank)
// ---------------------------------------------------------------------------
extern "C" void kernel_launch(void* const* d_in, const int* in_sizes, int n_in,
                              void* d_out, int out_size, void* d_ws, size_t ws_size,
                              hipStream_t stream) {
    const float* waveforms = (const float*)d_in[0];
    const float* low_hz    = (const float*)d_in[1];
    const float* band_hz   = (const float*)d_in[2];
    float* out  = (float*)d_out;
    float* filt = (float*)d_ws;

    build_filters_kernel<<<1, 64, 0, stream>>>(low_hz, band_hz, filt);

    dim3 grid(TOUT / TILE_T, 64 / NB);   // (127, 32)
    sinc_conv_wmma_kernel<<<grid, BLOCK, 0, stream>>>(waveforms, filt, out);
}